// AttentionOU_89902255440607
// MI455X (gfx1250) — compile-verified
//
#include <hip/hip_runtime.h>
#include <hip/hip_bf16.h>
#include <math.h>

typedef __attribute__((ext_vector_type(2))) float v2f;
typedef __attribute__((ext_vector_type(8))) float v8f;
typedef __attribute__((ext_vector_type(4))) unsigned int v4u;
typedef __attribute__((ext_vector_type(4))) int v4i;
typedef __attribute__((ext_vector_type(8))) int v8i;

#define BATCH 64
#define SEQ_S 256
#define NV 385
#define NQ 321
#define DIMC 768
#define NH 12
#define HD 64
#define QKV3 2304
#define TFL 64
#define ATTN_SCALE 0.125f

// workspace layout (float offsets)
#define OFF_TGT 0ull
#define OFF_H1  (OFF_TGT + (size_t)BATCH*DIMC)                 // B*768
#define OFF_H2  (OFF_H1  + (size_t)BATCH*SEQ_S*384)            // B*S*384
#define OFF_QKV (OFF_H2  + (size_t)BATCH*SEQ_S*192)            // B*S*192
#define OFF_AO  (OFF_QKV + (size_t)BATCH*NV*QKV3)              // B*Nv*2304
#define OFF_CLS (OFF_AO  + (size_t)BATCH*NQ*DIMC)              // B*Nq*768

__device__ __forceinline__ float gelu_exact(float v) {
    return 0.5f * v * (1.0f + erff(v * 0.70710678118654752f));
}

// ---------------------------------------------------------------------------
// TDM: async DMA of a rows x cols fp32 tile (row stride = stride_elems) from
// global into LDS, with LDS padding of 2 DWORDs every 64 DWORDs so the staged
// tile has a 66-float row pitch (bank-conflict-free, 8B-aligned rows).
// Descriptor per CDNA5 ISA. Tracked by TENSORcnt.
// ---------------------------------------------------------------------------
#define KC   64     // K-chunk (tile cols)
#define SROW 66     // LDS row pitch in floats after TDM padding

__device__ __forceinline__ void tdm_load_tile64(const float* gptr, void* lptr,
                                                unsigned stride_elems) {
    unsigned long long ga = (unsigned long long)(uintptr_t)gptr;
    unsigned lds = (unsigned)(uintptr_t)lptr;   // low 32 bits of generic = LDS addr
    const unsigned rows = 64u, cols = KC;

    v4u g0;
    g0.x = 1u;                                            // count=1, user desc
    g0.y = lds;                                           // [63:32] lds_addr
    g0.z = (unsigned)ga;                                  // [95:64] global lo
    g0.w = (unsigned)((ga >> 32) & 0x01FFFFFFull)         // [120:96] global hi
           | (2u << 30);                                  // [127:126] type=2

    v8i g1;
    g1[0] = (int)((2u << 16)          // data_size = 4B
                | (1u << 20)          // pad_enable
                | (5u << 22)          // pad_interval: every 64 DWORDs
                | (1u << 25));        // pad_amount: 2 DWORDs
    g1[1] = (int)((cols & 0xFFFFu) << 16);                    // tensor_dim0 lo
    g1[2] = (int)(((cols >> 16) & 0xFFFFu)
                | ((rows & 0xFFFFu) << 16));                  // dim0 hi | dim1 lo
    g1[3] = (int)(((rows >> 16) & 0xFFFFu)
                | ((cols & 0xFFFFu) << 16));                  // dim1 hi | tile_dim0
    g1[4] = (int)(rows & 0xFFFFu);                            // tile_dim1 (tile_dim2=0)
    g1[5] = (int)stride_elems;                                // tensor_dim0_stride lo
    g1[6] = 0;                                                // stride hi | dim1_stride lo
    g1[7] = 0;

    v4i z4 = {0, 0, 0, 0};
#if defined(__clang_major__) && (__clang_major__ >= 23)
    v8i z8 = {0, 0, 0, 0, 0, 0, 0, 0};
    __builtin_amdgcn_tensor_load_to_lds(g0, g1, z4, z4, z8, 0);
#else
    __builtin_amdgcn_tensor_load_to_lds(g0, g1, z4, z4, 0);
#endif
}

// ---- tgt_rep = mean over template rows 65..128 -------------------------------
__global__ void mean_kernel(const float* __restrict__ x, float* __restrict__ tgt) {
    int i = blockIdx.x * blockDim.x + threadIdx.x;
    if (i >= BATCH * DIMC) return;
    int b = i / DIMC, c = i % DIMC;
    const float* p = x + ((size_t)b * NV + 65) * DIMC + c;
    float s = 0.0f;
    for (int r = 0; r < 64; ++r) s += p[(size_t)r * DIMC];
    tgt[i] = s * (1.0f / 64.0f);
}

// ---------------------------------------------------------------------------
// fp32 WMMA GEMM with TDM-staged, double-buffered LDS tiles.
// C[M,N] = A[M,K] @ W[N,K]^T + bias (optional exact GELU).
// Block: 128 threads = 4 waves; block tile 64x64; wave w -> rows m0+w*16.
// Requires: M % 64 == 0, N % 64 == 0, K % 64 == 0 (true for all call sites).
// ---------------------------------------------------------------------------
__global__ void gemm_wmma(const float* __restrict__ A, const float* __restrict__ W,
                          const float* __restrict__ bias, float* __restrict__ C,
                          int M, int N, int K, int act) {
    __shared__ float sA[2][64 * SROW];
    __shared__ float sW[2][64 * SROW];

    int lane = threadIdx.x & 31;
    int wave = threadIdx.x >> 5;
    int n0 = blockIdx.x * 64;
    int m0 = blockIdx.y * 64;
    int mi = lane & 15;
    int koff = (lane >> 4) << 1;
    int nchunks = K / KC;

    if (wave == 0) {
        tdm_load_tile64(A + (size_t)m0 * K, &sA[0][0], (unsigned)K);
        tdm_load_tile64(W + (size_t)n0 * K, &sW[0][0], (unsigned)K);
    }

    v8f acc[4];
    v8f z8 = {0.f,0.f,0.f,0.f,0.f,0.f,0.f,0.f};
    #pragma unroll
    for (int j = 0; j < 4; ++j) acc[j] = z8;

    for (int c = 0; c < nchunks; ++c) {
        int buf = c & 1;
        if (wave == 0) {
            if (c + 1 < nchunks) {
                tdm_load_tile64(A + (size_t)m0 * K + (size_t)(c + 1) * KC,
                                &sA[buf ^ 1][0], (unsigned)K);
                tdm_load_tile64(W + (size_t)n0 * K + (size_t)(c + 1) * KC,
                                &sW[buf ^ 1][0], (unsigned)K);
                __builtin_amdgcn_s_wait_tensorcnt(2);  // chunk c fully landed
            } else {
                __builtin_amdgcn_s_wait_tensorcnt(0);
            }
        }
        __syncthreads();   // release staged chunk c to all waves

        const float* a_base = &sA[buf][(wave * 16 + mi) * SROW];
        #pragma unroll
        for (int k = 0; k < KC; k += 4) {
            int kk = k + koff;
            v2f a; a.x = a_base[kk]; a.y = a_base[kk + 1];
            #pragma unroll
            for (int j = 0; j < 4; ++j) {
                const float* w_base = &sW[buf][(j * 16 + mi) * SROW];
                v2f bb; bb.x = w_base[kk]; bb.y = w_base[kk + 1];
                acc[j] = __builtin_amdgcn_wmma_f32_16x16x4_f32(
                    false, a, false, bb, (short)0, acc[j], false, false);
            }
        }
        __syncthreads();   // chunk c consumed; its buffer may be overwritten
    }

    int mbase = m0 + wave * 16 + ((lane >> 4) << 3);
    #pragma unroll
    for (int j = 0; j < 4; ++j) {
        int n = n0 + j * 16 + mi;
        float bv = bias[n];
        #pragma unroll
        for (int r = 0; r < 8; ++r) {
            int m = mbase + r;
            float v = acc[j][r] + bv;
            if (act) v = gelu_exact(v);
            C[(size_t)m * N + n] = v;
        }
    }
}

// ---- dp1 GEMM with concat([x_search, tgt_rep]) A-operand, K=1536, N=384 ------
__global__ void gemm_dp1(const float* __restrict__ x, const float* __restrict__ tgt,
                         const float* __restrict__ W, const float* __restrict__ bias,
                         float* __restrict__ C) {
    const int N = 384, K = 1536;
    int lane = threadIdx.x & 31;
    int wave = threadIdx.x >> 5;
    int n0 = blockIdx.x * 64;
    int m0 = blockIdx.y * 64 + wave * 16;
    int mrow = m0 + (lane & 15);           // m in [0, B*S)
    int b = mrow >> 8, s = mrow & 255;
    const float* arow0 = x + ((size_t)(b * NV + 129 + s)) * DIMC;  // x[:, -S:]
    const float* arow1 = tgt + (size_t)b * DIMC;
    int koff = (lane >> 4) << 1;
    int ncol = n0 + (lane & 15);

    v8f acc[4];
    v8f z8 = {0.f,0.f,0.f,0.f,0.f,0.f,0.f,0.f};
    #pragma unroll
    for (int j = 0; j < 4; ++j) acc[j] = z8;

    for (int k = 0; k < K; k += 4) {
        int kk = k + koff;
        const float* src = (kk < DIMC) ? (arow0 + kk) : (arow1 + (kk - DIMC));
        v2f a; a.x = src[0]; a.y = src[1];
        #pragma unroll
        for (int j = 0; j < 4; ++j) {
            const float* wrow = W + (size_t)(ncol + j * 16) * K;
            v2f bb; bb.x = wrow[kk]; bb.y = wrow[kk + 1];
            acc[j] = __builtin_amdgcn_wmma_f32_16x16x4_f32(
                false, a, false, bb, (short)0, acc[j], false, false);
        }
    }

    int mbase = m0 + ((lane >> 4) << 3);
    #pragma unroll
    for (int j = 0; j < 4; ++j) {
        int n = n0 + j * 16 + (lane & 15);
        float bv = bias[n];
        #pragma unroll
        for (int r = 0; r < 8; ++r) {
            int m = mbase + r;
            float v = acc[j][r] + bv;
            C[(size_t)m * N + n] = gelu_exact(v);
        }
    }
}

// ---- per-row policy class: 2=token/p2, 0=template/p0, 1/2 for search rows ----
__global__ void cls_kernel(const float* __restrict__ h2, const float* __restrict__ w3,
                           const float* __restrict__ b3, int* __restrict__ cls) {
    int i = blockIdx.x * blockDim.x + threadIdx.x;
    if (i >= BATCH * NQ) return;
    int b = i / NQ, r = i % NQ;
    int c;
    if (r == 0) c = 2;
    else if (r <= 64) c = 0;
    else {
        int s = r - 65;
        const float* h = h2 + ((size_t)b * SEQ_S + s) * 192;
        float l0 = b3[0], l1 = b3[1];
        for (int k = 0; k < 192; ++k) { l0 += h[k] * w3[k]; l1 += h[k] * w3[192 + k]; }
        c = (l1 > l0) ? 2 : 1;   // argmax (ties -> index 0 -> class 1)
    }
    cls[i] = c;
}

// ---- attention: one wave per (b, head, 16-query tile) ------------------------
#define SPAD 401
__global__ void attn_kernel(const float* __restrict__ qkv, const int* __restrict__ cls,
                            float* __restrict__ out) {
    __shared__ float sP[16 * SPAD];
    int lane = threadIdx.x;
    int qt = blockIdx.x, h = blockIdx.y, b = blockIdx.z;
    int mi  = lane & 15;
    int hi2 = (lane >> 4) << 1;
    int qi = qt * 16 + mi;
    int vrow = (qi == 0) ? 0 : (qi + TFL);        // q rows: {0} U {65..384}
    if (qi >= NQ) vrow = 0;                       // pad rows: safe address, unstored

    // cache scaled Q fragments for K=64 (16 fragments of K=4)
    const float* qrow = qkv + ((size_t)(b * NV + vrow)) * QKV3 + h * HD;
    v2f aq[16];
    #pragma unroll
    for (int f = 0; f < 16; ++f) {
        int k = f * 4 + hi2;
        aq[f].x = qrow[k] * ATTN_SCALE;
        aq[f].y = qrow[k + 1] * ATTN_SCALE;
    }

    // scores = Q @ K^T -> LDS (16 x 400, cols >= 385 forced to -inf-ish)
    const float* kbase = qkv + (size_t)b * NV * QKV3 + DIMC + h * HD;
    for (int kt = 0; kt < 25; ++kt) {
        int kc = kt * 16 + mi;
        int kcv = kc < NV;
        const float* krow = kbase + (size_t)(kcv ? kc : 0) * QKV3;
        v8f acc = {0.f,0.f,0.f,0.f,0.f,0.f,0.f,0.f};
        #pragma unroll
        for (int f = 0; f < 16; ++f) {
            int k = f * 4 + hi2;
            v2f bb; bb.x = krow[k]; bb.y = krow[k + 1];
            acc = __builtin_amdgcn_wmma_f32_16x16x4_f32(
                false, aq[f], false, bb, (short)0, acc, false, false);
        }
        int col = kt * 16 + mi;
        int rbase = (lane >> 4) << 3;
        #pragma unroll
        for (int r = 0; r < 8; ++r)
            sP[(rbase + r) * SPAD + col] = kcv ? acc[r] : -1e30f;
    }
    __syncthreads();

    // masked softmax with reference's eps renormalization (lanes 0..15, 1 row each)
    if (lane < 16) {
        int row = lane;
        int qri = qt * 16 + row;
        int ci = (qri < NQ) ? cls[b * NQ + qri] : 0;
        float* sr = sP + row * SPAD;
        float rmax = -1e30f;
        for (int c2 = 0; c2 < NV; ++c2) rmax = fmaxf(rmax, sr[c2]);
        float sum = 0.0f;
        for (int c2 = 0; c2 < NV; ++c2) {
            int mcol = (c2 == 0) ? 0 : (c2 <= 64 ? c2 : c2 - 64);
            int cj = cls[b * NQ + mcol];
            int allowed = !((ci == 0 && cj == 1) || (ci == 1 && cj == 0));
            int mbit = allowed || (mcol == qri);  // + (1-mask)*eye
            float a = __expf(sr[c2] - rmax) * (float)mbit;
            sr[c2] = a;
            sum += a;
        }
        float inv = 1.0f / (sum + 1e-6f);
        const float epsq = 1e-6f / (float)NQ;
        for (int c2 = 0; c2 < NV; ++c2) sr[c2] = (sr[c2] + epsq) * inv;
        for (int c2 = NV; c2 < 400; ++c2) sr[c2] = 0.0f;
    }
    __syncthreads();

    // out = P @ V   (P: 16x400 in LDS, V: 385x64)
    const float* vbase = qkv + (size_t)b * NV * QKV3 + 2 * DIMC + h * HD;
    v8f po[4];
    v8f z8 = {0.f,0.f,0.f,0.f,0.f,0.f,0.f,0.f};
    #pragma unroll
    for (int j = 0; j < 4; ++j) po[j] = z8;
    for (int k = 0; k < 400; k += 4) {
        int k0 = k + hi2;
        v2f a; a.x = sP[mi * SPAD + k0]; a.y = sP[mi * SPAD + k0 + 1];
        int r0ok = k0 < NV, r1ok = (k0 + 1) < NV;
        const float* v0 = vbase + (size_t)(r0ok ? k0 : 0) * QKV3;
        const float* v1 = vbase + (size_t)(r1ok ? k0 + 1 : 0) * QKV3;
        #pragma unroll
        for (int j = 0; j < 4; ++j) {
            int n = j * 16 + mi;
            v2f bb;
            bb.x = r0ok ? v0[n] : 0.0f;
            bb.y = r1ok ? v1[n] : 0.0f;
            po[j] = __builtin_amdgcn_wmma_f32_16x16x4_f32(
                false, a, false, bb, (short)0, po[j], false, false);
        }
    }

    int rbase = (lane >> 4) << 3;
    #pragma unroll
    for (int j = 0; j < 4; ++j) {
        #pragma unroll
        for (int r = 0; r < 8; ++r) {
            int m = qt * 16 + rbase + r;
            if (m < NQ)
                out[((size_t)(b * NQ + m)) * DIMC + h * HD + j * 16 + mi] = po[j][r];
        }
    }
}

extern "C" void kernel_launch(void* const* d_in, const int* in_sizes, int n_in,
                              void* d_out, int out_size, void* d_ws, size_t ws_size,
                              hipStream_t stream) {
    const float* x      = (const float*)d_in[0];
    // d_in[1] = template_mask (unused in forward)
    const float* qkv_w  = (const float*)d_in[2];
    const float* qkv_b  = (const float*)d_in[3];
    const float* proj_w = (const float*)d_in[4];
    const float* proj_b = (const float*)d_in[5];
    const float* dp1_w  = (const float*)d_in[6];
    const float* dp1_b  = (const float*)d_in[7];
    const float* dp2_w  = (const float*)d_in[8];
    const float* dp2_b  = (const float*)d_in[9];
    const float* dp3_w  = (const float*)d_in[10];
    const float* dp3_b  = (const float*)d_in[11];

    float* ws   = (float*)d_ws;
    float* tgt  = ws + OFF_TGT;
    float* h1   = ws + OFF_H1;
    float* h2   = ws + OFF_H2;
    float* qkvb = ws + OFF_QKV;
    float* ao   = ws + OFF_AO;
    int*   cls  = (int*)(ws + OFF_CLS);
    float* out  = (float*)d_out;

    mean_kernel<<<(BATCH * DIMC + 255) / 256, 256, 0, stream>>>(x, tgt);

    // dp1: M=16384, N=384, K=1536, GELU
    gemm_dp1<<<dim3(384 / 64, (BATCH * SEQ_S) / 64), 128, 0, stream>>>(x, tgt, dp1_w, dp1_b, h1);

    // dp2: M=16384, N=192, K=384, GELU (TDM-staged)
    gemm_wmma<<<dim3(192 / 64, (BATCH * SEQ_S) / 64), 128, 0, stream>>>(
        h1, dp2_w, dp2_b, h2, BATCH * SEQ_S, 192, 384, 1);

    cls_kernel<<<(BATCH * NQ + 255) / 256, 256, 0, stream>>>(h2, dp3_w, dp3_b, cls);

    // qkv: M=24640, N=2304, K=768 (TDM-staged)
    gemm_wmma<<<dim3(QKV3 / 64, (BATCH * NV) / 64), 128, 0, stream>>>(
        x, qkv_w, qkv_b, qkvb, BATCH * NV, QKV3, DIMC, 0);

    // attention: 21 query tiles x 12 heads x 64 batches, one wave each
    attn_kernel<<<dim3(21, NH, BATCH), 32, 0, stream>>>(qkvb, cls, ao);

    // proj: M=20544, N=768, K=768 -> d_out (TDM-staged)
    gemm_wmma<<<dim3(DIMC / 64, (BATCH * NQ) / 64), 128, 0, stream>>>(
        ao, proj_w, proj_b, out, BATCH * NQ, DIMC, DIMC, 0);
}